// BeliefMatchingLoss_58239756533919
// MI455X (gfx1250) — compile-verified
//
#include <hip/hip_runtime.h>

// Problem constants (fixed by the reference).
constexpr int  kB    = 8;
constexpr int  kC    = 19;
constexpr int  kHW   = 512 * 512;          // 262144 = 2^18
constexpr int  kNPix = kB * kHW;           // 2097152
constexpr int  kPPT  = 4;                  // pixels per thread
constexpr int  kTPB  = 256;                // 8 waves (wave32)
constexpr int  kNBlk = kNPix / (kPPT * kTPB);  // 2048, exact
constexpr int  kIgnore = 255;
constexpr float kCoef  = 0.01f;

typedef float v2f __attribute__((ext_vector_type(2)));
typedef float v8f __attribute__((ext_vector_type(8)));

__device__ __forceinline__ float fast_ln(float x) {
  // v_log_f32 is log2; scale to natural log.
  return __builtin_amdgcn_logf(x) * 0.69314718055994531f;
}
__device__ __forceinline__ float fast_exp(float x) {
  // v_exp_f32 is exp2.
  return __builtin_amdgcn_exp2f(x * 1.44269504088896341f);
}

// Fused digamma + lgamma for a > 0 via shift-by-4 + asymptotic series.
// Shares ln(z), 1/z and the pole product between the two functions.
//   psi(a)    = psi(z) - P'(a)/P(a),  z = a+4, P = a(a+1)(a+2)(a+3)
//   lgamma(a) = lgamma(z) - ln(P)
// With u = a(a+3):  P = u(u+2),  P' = 2(u+1)(2a+3).
__device__ __forceinline__ void psi_lgamma(float a, float& dg, float& lg) {
  const float z   = a + 4.0f;
  const float rz  = __builtin_amdgcn_rcpf(z);
  const float lnz = fast_ln(z);
  const float rz2 = rz * rz;
  // psi(z) = ln z - 1/(2z) - 1/(12 z^2) + 1/(120 z^4) - 1/(252 z^6)
  const float psi_z = lnz - 0.5f * rz
      - rz2 * (0.0833333333f - rz2 * (0.00833333333f - rz2 * 0.00396825397f));
  // lgamma(z) = (z-1/2)ln z - z + ln(2*pi)/2 + 1/(12z) - 1/(360 z^3) + 1/(1260 z^5)
  const float lg_z = (z - 0.5f) * lnz - z + 0.91893853320467274f
      + rz * (0.0833333333f - rz2 * (0.00277777778f - rz2 * 7.93650794e-4f));
  const float u   = a * (a + 3.0f);
  const float P   = u * (u + 2.0f);
  const float S   = 2.0f * (u + 1.0f) * (2.0f * a + 3.0f) * __builtin_amdgcn_rcpf(P);
  dg = psi_z - S;
  lg = lg_z - fast_ln(P);
}

// Wave32 sum via V_WMMA_F32_16X16X4_F32.
// A (16x4 f32): lane L<16 -> VGPR0=(M=L,K=0), VGPR1=(M=L,K=1);
//               lane L>=16 -> VGPR0=(M=L-16,K=2), VGPR1=(M=L-16,K=3).
// Set A = [s, 0]; B = all-ones (layout-invariant); D rows = s_m + s_{m+16}.
// Sum of 8 D VGPRs per lane = sum over rows 0..7 (lanes<16) or 8..15 (lanes>=16);
// one xor-16 shuffle completes the 32-lane sum. Requires EXEC all-ones.
__device__ __forceinline__ float wave_sum32(float s) {
  v2f a;  a[0] = s;    a[1] = 0.0f;
  v2f bo; bo[0] = 1.0f; bo[1] = 1.0f;
  v8f c = {0.0f, 0.0f, 0.0f, 0.0f, 0.0f, 0.0f, 0.0f, 0.0f};
  v8f d = __builtin_amdgcn_wmma_f32_16x16x4_f32(
      /*neg_a=*/false, a, /*neg_b=*/false, bo,
      /*c_mod=*/(short)0, c, /*reuse_a=*/false, /*reuse_b=*/false);
  float t = ((d[0] + d[1]) + (d[2] + d[3])) + ((d[4] + d[5]) + (d[6] + d[7]));
  t += __shfl_xor(t, 16, 32);
  return t;  // full wave sum broadcast to all lanes
}

__global__ __launch_bounds__(kTPB)
void bml_main(const float* __restrict__ pred, const int* __restrict__ target,
              float* __restrict__ partial) {
  const int tid = blockIdx.x * kTPB + threadIdx.x;
  const int p0  = tid * kPPT;             // 4 consecutive pixels, never cross a batch
  const int b   = p0 >> 18;               // kHW = 2^18
  const int hw  = p0 & (kHW - 1);
  const float* pb = pred + b * (kC * kHW) + hw;

  const int4 tg = *(const int4*)(target + p0);
  int t[4] = {tg.x, tg.y, tg.z, tg.w};
  bool ign[4];
  int  tc[4];
#pragma unroll
  for (int i = 0; i < 4; ++i) {
    ign[i] = (t[i] == kIgnore);
    tc[i]  = ign[i] ? 0 : t[i];
  }

  float a0[4]  = {0.f, 0.f, 0.f, 0.f};
  float slg[4] = {0.f, 0.f, 0.f, 0.f};
  float sd[4]  = {0.f, 0.f, 0.f, 0.f};
  float dan[4] = {0.f, 0.f, 0.f, 0.f};

#pragma unroll 1
  for (int c = 0; c < kC; ++c) {
    const float4 x = *(const float4*)(pb + c * kHW);
    // Each channel step jumps 1 MB: prefetch two channels ahead (clamped;
    // re-prefetching the last line twice is harmless).
    const int cp = (c + 2 < kC) ? (c + 2) : (kC - 1);
    __builtin_prefetch(pb + cp * kHW, 0, 1);
    const float xv[4] = {x.x, x.y, x.z, x.w};
#pragma unroll
    for (int i = 0; i < 4; ++i) {
      const float al = fast_exp(xv[i]);
      float dg, lg;
      psi_lgamma(al, dg, lg);
      a0[i]  += al;
      slg[i] += lg;
      sd[i]   = __builtin_fmaf(al - 1.0f, dg, sd[i]);
      dan[i]  = (c == tc[i]) ? dg : dan[i];
    }
  }

  float loss_sum = 0.0f, cnt = 0.0f;
#pragma unroll
  for (int i = 0; i < 4; ++i) {
    float dg0, lg0;
    psi_lgamma(a0[i], dg0, lg0);
    const float kl   = lg0 - slg[i] + sd[i] - (a0[i] - (float)kC) * dg0;
    const float loss = kCoef * kl - (dan[i] - dg0);
    loss_sum += ign[i] ? 0.0f : loss;
    cnt      += ign[i] ? 0.0f : 1.0f;
  }

  // Wave32 reduction via WMMA (EXEC all-ones here: exact grid, no divergence).
  const float wsum = wave_sum32(loss_sum);
  const float wcnt = wave_sum32(cnt);

  __shared__ float red[16];
  const int lane = threadIdx.x & 31;
  const int wv   = threadIdx.x >> 5;
  if (lane == 0) { red[wv] = wsum; red[8 + wv] = wcnt; }
  __syncthreads();
  if (threadIdx.x == 0) {
    float bs = 0.0f, bc = 0.0f;
#pragma unroll
    for (int i = 0; i < 8; ++i) { bs += red[i]; bc += red[8 + i]; }
    partial[blockIdx.x]         = bs;   // every slot rewritten each launch
    partial[kNBlk + blockIdx.x] = bc;
  }
}

__global__ __launch_bounds__(256)
void bml_finalize(const float* __restrict__ partial, float* __restrict__ out) {
  __shared__ double sh[512];
  double s = 0.0, c = 0.0;
  for (int i = threadIdx.x; i < kNBlk; i += 256) {
    s += (double)partial[i];
    c += (double)partial[kNBlk + i];
  }
  sh[threadIdx.x]       = s;
  sh[256 + threadIdx.x] = c;
  __syncthreads();
  for (int off = 128; off > 0; off >>= 1) {
    if (threadIdx.x < off) {
      sh[threadIdx.x]       += sh[threadIdx.x + off];
      sh[256 + threadIdx.x] += sh[256 + threadIdx.x + off];
    }
    __syncthreads();
  }
  if (threadIdx.x == 0) out[0] = (float)(sh[0] / sh[256]);
}

extern "C" void kernel_launch(void* const* d_in, const int* in_sizes, int n_in,
                              void* d_out, int out_size, void* d_ws, size_t ws_size,
                              hipStream_t stream) {
  const float* pred   = (const float*)d_in[0];
  const int*   target = (const int*)d_in[1];
  float*       out    = (float*)d_out;
  float*       part   = (float*)d_ws;  // 2*kNBlk floats = 16 KB

  bml_main<<<kNBlk, kTPB, 0, stream>>>(pred, target, part);
  bml_finalize<<<1, 256, 0, stream>>>(part, out);
}